// SAFM_2473901162526
// MI455X (gfx1250) — compile-verified
//
#include <hip/hip_runtime.h>
#include <hip/hip_bf16.h>

typedef float v2f __attribute__((ext_vector_type(2)));
typedef float v8f __attribute__((ext_vector_type(8)));

#define CCH   64            // channels
#define HW    65536         // 256*256 pixels per image
#define BATCH 8
#define NT_TOTAL ((BATCH * HW) / 16)   // 32768 16-pixel tiles

// out[b,o,p] = sum_c w[o,c] * x[b,c,p] + bias[o]   (everything else in the
// reference is dead: the channel permutation is exactly inverted, and the
// FFT/GELU branches only feed an unused tensor).
//
// Wave pair (2w, 2w+1) shares one 16-pixel tile; each wave computes 32 of the
// 64 output channels as 16 K-steps x 2 o-tiles of V_WMMA_F32_16X16X4_F32.
__global__ __launch_bounds__(256)
void SAFM_refine_wmma(const float* __restrict__ x,
                      const float* __restrict__ w,
                      const float* __restrict__ bias,
                      float* __restrict__ out) {
    const int lane = threadIdx.x & 31;
    const int l    = lane & 15;     // M row (A) / N col (B,C,D)
    const int half = lane >> 4;     // K-half select

    // Wave-uniform values forced into SGPRs so all bases are scalar and
    // loads/stores take saddr-form with shared per-lane voffsets.
    const int wib    = __builtin_amdgcn_readfirstlane((int)(threadIdx.x >> 5));
    const int waveId = (int)blockIdx.x * 8 + wib;     // 8 wave32 per block
    const int nWaves = (int)gridDim.x * 8;
    const int owave  = waveId & 1;                    // which 32-channel half
    const int tw     = waveId >> 1;                   // tile worker id
    const int tStride = nWaves >> 1;

    const size_t CHW = (size_t)CCH * HW;

    // ---- Preload A (weights) for this wave's 32 output channels: 64 VGPRs.
    // A 16x4 f32 layout: lanes 0-15 M rows; VGPR0 = K0|K2, VGPR1 = K1|K3.
    const float* wBase = w + (size_t)(owave * 32) * CCH;   // scalar base
    const int voffA = l * CCH + half * 2;                  // per-lane
    v2f wA[2][16];
#pragma unroll
    for (int ot = 0; ot < 2; ++ot) {
#pragma unroll
        for (int kk = 0; kk < 16; ++kk) {
            wA[ot][kk] = *(const v2f*)(wBase + voffA + (ot * 16 * CCH + kk * 4));
        }
    }

    // ---- Per-lane bias for accumulator init: C/D VGPR r holds M = r + 8*half.
    const float* bBase = bias + owave * 32;                // scalar base
    float biasv[2][8];
#pragma unroll
    for (int ot = 0; ot < 2; ++ot) {
#pragma unroll
        for (int r = 0; r < 8; ++r) {
            biasv[ot][r] = bBase[8 * half + (ot * 16 + r)];
        }
    }

    // Shared per-lane voffsets (elements).
    const int voffB = half * 2 * HW + l;   // B: K-half rows
    const int voffS = half * 8 * HW + l;   // store: M = r + 8*half

    // ---- Scalar-bounded grid-stride loop (EXEC stays all-1s for WMMA).
    for (int t = tw; t < NT_TOTAL; t += tStride) {
        const int b  = t >> 12;            // t / (HW/16)
        const int p0 = (t & 4095) << 4;    // pixel offset within image

        // Bases centered so all channel-row constants fit signed 24-bit imms.
        const float* xb32 = x   + (size_t)b * CHW + p0 + 32 * HW;
        float*       obW  = out + (size_t)b * CHW + p0 + (size_t)(owave * 32 + 16) * HW;

        // Prefetch next tile's x (two cachelines per lane, rows lane and lane+32, cover all 64 rows).
        const int tn = t + tStride;
        if (tn < NT_TOTAL) {
            const float* xp = x + (size_t)(tn >> 12) * CHW + ((tn & 4095) << 4);
            __builtin_prefetch(xp + (size_t)lane * HW, 0, 3);
            __builtin_prefetch(xp + (size_t)(lane + 32) * HW, 0, 3);
        }

        v8f acc[2];
#pragma unroll
        for (int ot = 0; ot < 2; ++ot) {
#pragma unroll
            for (int r = 0; r < 8; ++r) acc[ot][r] = biasv[ot][r];
        }

#pragma unroll
        for (int kk = 0; kk < 16; ++kk) {
            // B 4x16 tile rows c0..c0+3; imm range (kk*4-32)*HW*4 in [-8388608, 7602176].
            const float* xr = xb32 + voffB + (kk * 4 - 32) * HW;
            v2f bm;
            bm.x = xr[0];       // K = c0 + 2*half
            bm.y = xr[HW];      // K = c0 + 2*half + 1
#pragma unroll
            for (int ot = 0; ot < 2; ++ot) {
                acc[ot] = __builtin_amdgcn_wmma_f32_16x16x4_f32(
                    false, wA[ot][kk], false, bm, (short)0, acc[ot], false, false);
            }
        }

        // ---- Store: o = owave*32 + ot*16 + r + 8*half; imm (ot*16+r-16)*HW*4.
#pragma unroll
        for (int ot = 0; ot < 2; ++ot) {
#pragma unroll
            for (int r = 0; r < 8; ++r) {
                obW[voffS + (ot * 16 + r - 16) * HW] = acc[ot][r];
            }
        }
    }
}

extern "C" void kernel_launch(void* const* d_in, const int* in_sizes, int n_in,
                              void* d_out, int out_size, void* d_ws, size_t ws_size,
                              hipStream_t stream) {
    // setup_inputs order: 0:x ... 11:refine_w 12:refine_b
    const float* x   = (const float*)d_in[0];
    const float* rw  = (const float*)d_in[11];
    const float* rb  = (const float*)d_in[12];
    float*       out = (float*)d_out;

    const int threads = 256;   // 8 wave32 per block
    const int blocks  = 2048;  // 16384 waves -> 8192 tile workers -> 4 tiles each
    SAFM_refine_wmma<<<blocks, threads, 0, stream>>>(x, rw, rb, out);
}